// deform_conv_model_60962765799489
// MI455X (gfx1250) — compile-verified
//
#include <hip/hip_runtime.h>

typedef __attribute__((ext_vector_type(2))) float v2f;
typedef __attribute__((ext_vector_type(8))) float v8f;

#define CIN   128
#define COUT  128
#define IMH   128
#define IMW   128
#define HWSZ  (IMH * IMW)          // 16384
#define CHW   (CIN * HWSZ)         // 2097152
#define HO    64
#define WO    64
#define NPIX  (HO * WO)            // 4096
#define KPTS  9
#define KCDIM (CIN * KPTS)         // 1152
#define TN    32                   // pixels per tile (two 16-wide B tiles)
#define NB    8

#define TILE_FLOATS (KCDIM * 16)        // 18432 floats per 16-pixel val tile
#define TILE_STRIDE (TILE_FLOATS + 16)  // +16 floats: shift banks by 16
#define NPAIR (KPTS * TN)               // 288 (pixel, tap) pairs

// Dynamic LDS layout (floats):
//   sVal : 2 * TILE_STRIDE           = 36896 floats (147584 B)
//   sOff : NPAIR*4 ints              =  4608 B
//   sW   : NPAIR*4 floats            =  4608 B
#define SMEM_BYTES ((2 * TILE_STRIDE + NPAIR * 4 * 2) * 4)  // 156800 B

__global__ __launch_bounds__(256) void dcn_transpose_weight(
    const float* __restrict__ w, float* __restrict__ wT) {
  int o = blockIdx.x * 256 + threadIdx.x;  // o = kc*COUT + co
  if (o < KCDIM * COUT) {
    int co = o & (COUT - 1);
    int kc = o >> 7;
    wT[o] = w[co * KCDIM + kc];
  }
}

__global__ __launch_bounds__(256) void dcn_main(
    const float* __restrict__ x, const float* __restrict__ offset,
    const float* __restrict__ mask, const float* __restrict__ wmat,
    const float* __restrict__ bias, float* __restrict__ out, int wTransposed) {
  extern __shared__ char smem_raw[];
  float* sVal = (float*)smem_raw;                            // 2 x [KCDIM][16]
  int*   sOff = (int*)(smem_raw + 2 * TILE_STRIDE * 4);      // [NPAIR][4]
  float* sW   = (float*)(smem_raw + 2 * TILE_STRIDE * 4 + NPAIR * 16);

  const int tid     = threadIdx.x;
  const int tile    = blockIdx.x;               // 1024 tiles
  const int pBase   = tile * TN;
  const int b       = pBase >> 12;              // / NPIX
  const int pixBase = pBase & (NPIX - 1);

  // ---------------- Phase 1: sampling parameters (32 pixels x 9 taps) ------
  for (int q = tid; q < NPAIR; q += 256) {
    int k   = q >> 5;             // tap 0..8
    int n   = q & 31;             // pixel in tile
    int pix = pixBase + n;
    int oy  = pix >> 6, ox = pix & 63;
    int ki  = k / 3, kj = k - 3 * ki;
    float gy = (float)(oy * 2 - 1 + ki);
    float gx = (float)(ox * 2 - 1 + kj);
    const float* offb = offset + (size_t)b * 2 * KPTS * NPIX;
    float dy = offb[(2 * k + 0) * NPIX + pix];
    float dx = offb[(2 * k + 1) * NPIX + pix];
    float py = dy + gy, px = dx + gx;
    float y0f = floorf(py), x0f = floorf(px);
    float fy = py - y0f, fx = px - x0f;
    float m = mask[((size_t)b * KPTS + k) * NPIX + pix];
    int y0 = (int)y0f, x0 = (int)x0f;
    float wv[4];
    wv[0] = (1.f - fy) * (1.f - fx) * m;
    wv[1] = (1.f - fy) * fx * m;
    wv[2] = fy * (1.f - fx) * m;
    wv[3] = fy * fx * m;
#pragma unroll
    for (int c4 = 0; c4 < 4; ++c4) {
      int yy = y0 + (c4 >> 1);
      int xx = x0 + (c4 & 1);
      bool valid = (yy >= 0) && (yy < IMH) && (xx >= 0) && (xx < IMW);
      int yc = yy < 0 ? 0 : (yy > IMH - 1 ? IMH - 1 : yy);
      int xc = xx < 0 ? 0 : (xx > IMW - 1 ? IMW - 1 : xx);
      sOff[q * 4 + c4] = yc * IMW + xc;
      sW[q * 4 + c4]   = valid ? wv[c4] : 0.f;
    }
  }
  __syncthreads();

  // ---------------- Phase 2: build val tiles [2][KCDIM][16] in LDS ---------
  {
    const float* xb = x + (size_t)b * CHW;
    for (int e = tid; e < KCDIM * TN; e += 256) {
      int c = e / NPAIR;
      int r = e - c * NPAIR;             // r = k*32 + n, matches sOff rows
      const float* xc = xb + c * HWSZ;
      int   o0 = sOff[r * 4 + 0], o1 = sOff[r * 4 + 1];
      int   o2 = sOff[r * 4 + 2], o3 = sOff[r * 4 + 3];
      float v = sW[r * 4 + 0] * xc[o0] + sW[r * 4 + 1] * xc[o1] +
                sW[r * 4 + 2] * xc[o2] + sW[r * 4 + 3] * xc[o3];
      int k = r >> 5, n = r & 31;
      int t = n >> 4, n16 = n & 15;
      sVal[t * TILE_STRIDE + (c * KPTS + k) * 16 + n16] = v;
    }
  }
  __syncthreads();

  // ---------------- Phase 3: GEMM via V_WMMA_F32_16X16X4_F32 ---------------
  const int wave = tid >> 5;
  const int lane = tid & 31;
  const int half = lane >> 4;  // 0: K = kc,kc+1   1: K = kc+2,kc+3
  const int lid  = lane & 15;  // A row / B col
  const int co0  = wave * 16;

  // Per-lane LDS bases for the two B tiles (includes the half's K shift).
  const float* sB0 = sVal + (2 * half) * 16 + lid;
  const float* sB1 = sB0 + TILE_STRIDE;

  v8f acc0 = {};
  v8f acc1 = {};

  if (wTransposed) {
    // wT[ka*COUT + co]: constant stride COUT -> immediate-offset loads.
    const float* wb = wmat + (size_t)(2 * half) * COUT + co0 + lid;
#pragma unroll 8
    for (int kc = 0; kc < KCDIM; kc += 4) {
      v2f a, b0, b1;
      a.x  = wb[(size_t)kc * COUT];
      a.y  = wb[(size_t)kc * COUT + COUT];
      b0.x = sB0[kc * 16];
      b0.y = sB0[kc * 16 + 16];
      b1.x = sB1[kc * 16];
      b1.y = sB1[kc * 16 + 16];
      acc0 = __builtin_amdgcn_wmma_f32_16x16x4_f32(false, a, false, b0,
                                                   (short)0, acc0, false, false);
      acc1 = __builtin_amdgcn_wmma_f32_16x16x4_f32(false, a, false, b1,
                                                   (short)0, acc1, false, false);
    }
  } else {
    // Original layout w[co*KCDIM + ka]: the lane's two K values are adjacent.
    const float* wrow = wmat + (size_t)(co0 + lid) * KCDIM + 2 * half;
#pragma unroll 8
    for (int kc = 0; kc < KCDIM; kc += 4) {
      v2f a, b0, b1;
      a.x  = wrow[kc];
      a.y  = wrow[kc + 1];
      b0.x = sB0[kc * 16];
      b0.y = sB0[kc * 16 + 16];
      b1.x = sB1[kc * 16];
      b1.y = sB1[kc * 16 + 16];
      acc0 = __builtin_amdgcn_wmma_f32_16x16x4_f32(false, a, false, b0,
                                                   (short)0, acc0, false, false);
      acc1 = __builtin_amdgcn_wmma_f32_16x16x4_f32(false, a, false, b1,
                                                   (short)0, acc1, false, false);
    }
  }

  // ---------------- Epilogue: bias + coalesced stores ----------------------
  float* outb = out + (size_t)b * COUT * NPIX;
#pragma unroll
  for (int vi = 0; vi < 8; ++vi) {
    int row = vi + 8 * half;  // C/D layout: row = vreg + 8*half, col = lid
    int co  = co0 + row;
    float bv = bias[co];
    float* orow = outb + (size_t)co * NPIX + pixBase;
    orow[lid]      = acc0[vi] + bv;
    orow[16 + lid] = acc1[vi] + bv;
  }
}

extern "C" void kernel_launch(void* const* d_in, const int* in_sizes, int n_in,
                              void* d_out, int out_size, void* d_ws,
                              size_t ws_size, hipStream_t stream) {
  const float* x      = (const float*)d_in[0];
  const float* offset = (const float*)d_in[1];
  const float* mask   = (const float*)d_in[2];
  const float* weight = (const float*)d_in[3];
  const float* bias   = (const float*)d_in[4];
  float* out = (float*)d_out;

  const size_t wT_bytes = (size_t)KCDIM * COUT * sizeof(float);  // 589824
  int useT = (d_ws != nullptr && ws_size >= wT_bytes) ? 1 : 0;
  const float* wptr = weight;
  if (useT) {
    float* wT = (float*)d_ws;
    dcn_transpose_weight<<<(KCDIM * COUT + 255) / 256, 256, 0, stream>>>(weight,
                                                                         wT);
    wptr = wT;
  }

  const int ntiles = NB * NPIX / TN;  // 1024
  dcn_main<<<ntiles, 256, SMEM_BYTES, stream>>>(x, offset, mask, wptr, bias,
                                                out, useT);
}